// HouseholderModule_79611513798825
// MI455X (gfx1250) — compile-verified
//
#include <hip/hip_runtime.h>

// Householder 'data_qk': out[b,h,l,:] = x[b,h,l,:] - 2*(x.v_h)*v_h,  v_h = normalize(R0[h])
// x: [4, 32, 4096, 128] f32, R0: [32, 128] f32, out like x.
// Memory-bound: 537 MB traffic -> ~23 us floor @ 23.3 TB/s. Rank-1 form: ~0.4 GFLOP.

#define NB    4
#define NH    32
#define NL    4096
#define ND    128
#define PITCH 132              // LDS floats per row: 128 + 4 pad -> conflict-free banks
#define TILE_ROWS 16
#define WAVES 4
#define TILES_PER_WAVE 4
#define LDS_TILE_FLOATS (TILE_ROWS * PITCH)    // 2112
#define LDS_TILE_BYTES  (LDS_TILE_FLOATS * 4)  // 8448

typedef __attribute__((ext_vector_type(2))) float v2f;
typedef __attribute__((ext_vector_type(4))) float v4f;
typedef __attribute__((ext_vector_type(8))) float v8f;

__device__ __forceinline__ float readlane_f(float x, int lane) {
  return __builtin_bit_cast(float, __builtin_amdgcn_readlane(__builtin_bit_cast(int, x), lane));
}

// CDNA5 async copy: one 512B x-row per instruction, tracked on ASYNCcnt.
// vdst = per-lane LDS byte offset, vaddr = per-lane 64-bit global address.
// x is read exactly once -> non-temporal so the 268MB stream doesn't displace L2.
__device__ __forceinline__ void async_row_b128(const float* g, unsigned lds_byte_off) {
  unsigned long long ga = (unsigned long long)(uintptr_t)g;
  asm volatile("global_load_async_to_lds_b128 %0, %1, off th:TH_LOAD_NT"
               :: "v"(lds_byte_off), "v"(ga)
               : "memory");
}

__device__ __forceinline__ void wait_async_le16() {
  asm volatile("s_wait_asynccnt 0x10" ::: "memory");
}
__device__ __forceinline__ void wait_async_0() {
  asm volatile("s_wait_asynccnt 0x0" ::: "memory");
}

__global__ void __launch_bounds__(WAVES * 32)
householder_qk_kernel(const float* __restrict__ x,
                      const float* __restrict__ R0,
                      float* __restrict__ out) {
  extern __shared__ float smem[];

  const int tid  = threadIdx.x;
  const int wave = tid >> 5;
  const int lane = tid & 31;
  const int bh   = blockIdx.x;        // 0..127 (b*32 + h)
  const int h    = bh & (NH - 1);

  // ---- per-wave unit Householder vector: lane l holds v[4l..4l+3] ----
  v4f r0 = *(const v4f*)(R0 + h * ND + 4 * lane);
  float ss = r0.x * r0.x + r0.y * r0.y + r0.z * r0.z + r0.w * r0.w;
#pragma unroll
  for (int off = 16; off > 0; off >>= 1) ss += __shfl_xor(ss, off, 32);
  const float inv = 1.0f / sqrtf(ss);
  const v4f v = r0 * inv;

  // LDS layout: [WAVES][2][TILE] double buffers, then [WAVES][128] v strips.
  const unsigned ldsBase = (unsigned)(wave * 2 * LDS_TILE_BYTES);
  const float*   buf0    = smem + wave * 2 * LDS_TILE_FLOATS;
  float*         vbuf    = smem + WAVES * 2 * LDS_TILE_FLOATS + wave * ND;

  // stage v into this wave's LDS strip (broadcast source for WMMA B operands)
  *(v4f*)(vbuf + 4 * lane) = v;

  const int    tile0 = blockIdx.y * (WAVES * TILES_PER_WAVE) + wave * TILES_PER_WAVE;
  const size_t slab  = (size_t)bh * NL * ND;

  // prefetch tile 0 -> buffer 0 (16 async b128 rows, padded LDS pitch)
  {
    const float* g = x + slab + (size_t)tile0 * TILE_ROWS * ND;
#pragma unroll
    for (int r = 0; r < TILE_ROWS; ++r)
      async_row_b128(g + r * ND + 4 * lane, ldsBase + r * (PITCH * 4) + lane * 16);
  }

  const bool hiHalf = lane >= 16;
  const int  m      = lane & 15;

#pragma unroll
  for (int i = 0; i < TILES_PER_WAVE; ++i) {
    if (i + 1 < TILES_PER_WAVE) {
      // prefetch next tile into the other buffer, then wait for current tile only
      const float*   g       = x + slab + (size_t)(tile0 + i + 1) * TILE_ROWS * ND;
      const unsigned dstBase = ldsBase + (unsigned)(((i + 1) & 1) * LDS_TILE_BYTES);
#pragma unroll
      for (int r = 0; r < TILE_ROWS; ++r)
        async_row_b128(g + r * ND + 4 * lane, dstBase + r * (PITCH * 4) + lane * 16);
      wait_async_le16();
    } else {
      wait_async_0();
    }
    const float* buf = buf0 + (i & 1) * LDS_TILE_FLOATS;

    // ---- t[m] = dot(x[m], v) for 16 rows at once via v_wmma_f32_16x16x4_f32 ----
    // A (16x4, ISA layout): lanes 0-15 supply {K0,K1} of row m, lanes 16-31 {K2,K3}.
    // B (4x16): v[k0+k] broadcast to all 16 columns (ds_load_b64 from the v strip;
    //           one address per half-wave -> LDS broadcast, no EXEC games, no readlanes).
    v8f acc = {};  // C = 0
#pragma unroll
    for (int c = 0; c < ND / 4; ++c) {
      v2f a = *(const v2f*)(buf  + m * PITCH + 4 * c + (hiHalf ? 2 : 0));
      v2f b = *(const v2f*)(vbuf + 4 * c            + (hiHalf ? 2 : 0));
      acc = __builtin_amdgcn_wmma_f32_16x16x4_f32(
          /*neg_a=*/false, a, /*neg_b=*/false, b,
          /*c_mod=*/(short)0, acc, /*reuse_a=*/false, /*reuse_b=*/false);
    }

    // ---- out[r] = x[r] + (-2*t[r]) * v ;  D layout: VGPR j = rows j (lanes0-15) / j+8 (16-31)
    float* o = out + slab + (size_t)(tile0 + i) * TILE_ROWS * ND;
#pragma unroll
    for (int r = 0; r < TILE_ROWS; ++r) {
      float t = (r < 8) ? readlane_f(acc[r], 0) : readlane_f(acc[r - 8], 16);
      float s = -2.0f * t;
      v4f xr = *(const v4f*)(buf + r * PITCH + 4 * lane);
      v4f yr;
      yr.x = __builtin_fmaf(s, v.x, xr.x);
      yr.y = __builtin_fmaf(s, v.y, xr.y);
      yr.z = __builtin_fmaf(s, v.z, xr.z);
      yr.w = __builtin_fmaf(s, v.w, xr.w);
      __builtin_nontemporal_store(yr, (v4f*)(o + r * ND + 4 * lane));
    }
  }
}

extern "C" void kernel_launch(void* const* d_in, const int* in_sizes, int n_in,
                              void* d_out, int out_size, void* d_ws, size_t ws_size,
                              hipStream_t stream) {
  const float* x  = (const float*)d_in[0];
  const float* R0 = (const float*)d_in[1];
  float*       o  = (float*)d_out;

  dim3 grid(NB * NH, NL / (TILE_ROWS * WAVES * TILES_PER_WAVE));   // (128, 16)
  dim3 block(WAVES * 32);                                           // 128 threads, 4 waves
  size_t smem = (size_t)(WAVES * 2 * LDS_TILE_FLOATS + WAVES * ND) * sizeof(float); // 69632 B

  hipLaunchKernelGGL(householder_qk_kernel, grid, block, smem, stream, x, R0, o);
}